// SINDy_order2_29111288332796
// MI455X (gfx1250) — compile-verified
//
#include <hip/hip_runtime.h>
#include <stdint.h>

// SINDy order-3 feature library: out[b, 0:985] =
//   [1, zc(16), zc_i*zc_j (136, i<=j), zc_i*zc_j*zc_k (816, i<=j<=k), sin(zc)(16)]
// Store-bandwidth bound: 516.6 MB out vs 8.4 MB in -> ~22us floor at 23.3 TB/s.

#define NZ              16
#define ROWS_PER_BLOCK  8
#define ROW_STRIDE      20      // floats (80B): keeps 16B alignment for B128 async chunks
#define NFEAT           985
#define NITER_MAIN      30      // f in [0, 960): pure products, no guards
#define TAB_PAD         992     // 31*32, padded so last iteration needs no table guard

struct FeatTab { unsigned v[TAB_PAD]; };

static constexpr unsigned pack_idx(int i, int j, int k, unsigned flag) {
  return (unsigned)i | ((unsigned)j << 8) | ((unsigned)k << 16) | (flag << 24);
}

static constexpr FeatTab make_tab() {
  FeatTab t{};
  int f = 0;
  // slot 16 of the per-row LDS vector holds 1.0f, so every non-sin feature is a
  // uniform 3-factor product.
  t.v[f++] = pack_idx(16, 16, 16, 0);                              // ones
  for (int i = 0; i < NZ; ++i) t.v[f++] = pack_idx(i, 16, 16, 0);  // linear
  for (int i = 0; i < NZ; ++i)                                      // deg2, i<=j
    for (int j = i; j < NZ; ++j) t.v[f++] = pack_idx(i, j, 16, 0);
  for (int i = 0; i < NZ; ++i)                                      // deg3, i<=j<=k
    for (int j = i; j < NZ; ++j)
      for (int k = j; k < NZ; ++k) t.v[f++] = pack_idx(i, j, k, 0);
  for (int i = 0; i < NZ; ++i) t.v[f++] = pack_idx(i, 16, 16, 1);   // sin(zc_i)
  while (f < TAB_PAD) t.v[f++] = pack_idx(16, 16, 16, 0);           // pad (never stored)
  return t;
}

__constant__ FeatTab g_tab = make_tab();

__device__ __forceinline__ float bperm_f32(unsigned idx, int vbits) {
  // wave32 lane gather: dst[lane] = src[idx]; idx is a lane id (shifted to a byte address), bits [6:2] used.
  return __int_as_float(__builtin_amdgcn_ds_bpermute((int)(idx << 2), vbits));
}

__global__ __launch_bounds__(256) void sindy_feat_kernel(
    const float* __restrict__ z, const float* __restrict__ dz,
    float* __restrict__ out, int nrows) {
  __shared__ float smem[ROWS_PER_BLOCK * ROW_STRIDE];

  const int t    = threadIdx.x;
  const int lane = t & 31;
  const int wv   = t >> 5;
  const long long row0 = (long long)blockIdx.x * ROWS_PER_BLOCK;

  // Row padding value: product identity for the packed index 16.
  if (t < ROWS_PER_BLOCK) smem[t * ROW_STRIDE + NZ] = 1.0f;

  // Stage 8 rows x (8 z + 8 dz) floats = 32 x 16B chunks via the gfx1250
  // async global->LDS path (ASYNCcnt). Wave 0 issues one B128 per lane.
  if (t < 32) {
    const int rl = t >> 2;        // local row 0..7
    const int part = t & 3;       // 0,1 -> z halves; 2,3 -> dz halves
    const long long row = row0 + rl;
    if (row < nrows) {
      const char* base = (part < 2) ? (const char*)z : (const char*)dz;
      const unsigned long long gaddr =
          (unsigned long long)(uintptr_t)(base + row * 32 + (long long)(part & 1) * 16);
      const unsigned laddr =
          (unsigned)(uintptr_t)(&smem[rl * ROW_STRIDE]) + (unsigned)(part & 1) * 16u
          + ((part < 2) ? 0u : 32u);
      asm volatile("global_load_async_to_lds_b128 %0, %1, off"
                   :: "v"(laddr), "v"(gaddr)
                   : "memory");
    }
  }
  asm volatile("s_wait_asynccnt 0" ::: "memory");
  __syncthreads();

  const long long row = row0 + wv;          // one wave per output row
  if (row >= nrows) return;

  float vself = (lane <= NZ) ? smem[wv * ROW_STRIDE + lane] : 0.0f;
  const int vbits = __float_as_int(vself);

  float* __restrict__ orow = out + row * (long long)NFEAT;

  // f in [0, 960): all product features. Coalesced 128B/wave NT stores.
#pragma unroll
  for (int it = 0; it < NITER_MAIN; ++it) {
    const int f = it * 32 + lane;
    const unsigned u = g_tab.v[f];
    const float a = bperm_f32(u & 255u, vbits);
    const float b = bperm_f32((u >> 8) & 255u, vbits);
    const float c = bperm_f32((u >> 16) & 255u, vbits);
    __builtin_nontemporal_store(a * b * c, orow + f);
  }

  // Tail iteration: f in [960, 992). 960..968 = last deg3 products,
  // 969..984 = sin(zc), 985..991 = padding (not stored). All bpermutes run
  // with full EXEC (required: disabled source lanes read as zero); only the
  // cheap VALU sin body and the store are predicated.
  {
    const int f = NITER_MAIN * 32 + lane;
    const unsigned u = g_tab.v[f];
    const float a = bperm_f32(u & 255u, vbits);
    const float b = bperm_f32((u >> 8) & 255u, vbits);
    const float c = bperm_f32((u >> 16) & 255u, vbits);
    float r = a * b * c;
    if (u >> 24) r = sinf(a);     // single sin call-site in the whole kernel
    if (f < NFEAT) __builtin_nontemporal_store(r, orow + f);
  }
}

extern "C" void kernel_launch(void* const* d_in, const int* in_sizes, int n_in,
                              void* d_out, int out_size, void* d_ws, size_t ws_size,
                              hipStream_t stream) {
  (void)n_in; (void)out_size; (void)d_ws; (void)ws_size;
  const float* z  = (const float*)d_in[0];   // [B, 8] f32
  const float* dz = (const float*)d_in[1];   // [B, 8] f32
  float* out = (float*)d_out;                // [B, 985] f32
  const int nrows = in_sizes[0] / 8;
  const int nblocks = (nrows + ROWS_PER_BLOCK - 1) / ROWS_PER_BLOCK;
  sindy_feat_kernel<<<nblocks, 256, 0, stream>>>(z, dz, out, nrows);
}